// PointTransformerSeg_45578192945727
// MI455X (gfx1250) — compile-verified
//
#include <hip/hip_runtime.h>
#include <hip/hip_bf16.h>
#include <math.h>

// ---------------------------------------------------------------------------
// Point Transformer forward on gfx1250 (MI455X).
// Dense channel contractions run through V_WMMA_F32_16X16X4_F32 (fp32 WMMA,
// wave32). Each wave computes a 32x(16*NSUB) output tile: 2 A fragments x
// NSUB B fragments -> 2*NSUB independent WMMA chains per K-step (12 dword
// loads per 8 WMMAs at NSUB=4). The K-loop is software-pipelined (next
// fragments are in flight while current WMMAs execute) and all operand
// offsets are 32-bit so the backend can use saddr+voffset addressing.
// Bounds handled by clamped addressing; out-of-range lanes masked at store.
//
// Parameter order assumption: d_in[0] = pc; d_in[1..] = params leaves in
// jax tree-flatten order (dict keys sorted recursively, lists in order).
// ---------------------------------------------------------------------------

typedef float v2f __attribute__((ext_vector_type(2)));
typedef float v8f __attribute__((ext_vector_type(8)));

#define BATCH 2
#define GEMM_WAVES 4

#define WMMA_F32(a, b, c) __builtin_amdgcn_wmma_f32_16x16x4_f32( \
    false, (a), false, (b), (short)0, (c), false, false)

// ======================= WMMA GEMM (fp32, 16x16x4) =========================
// C[M,N] = op(A[M,K](lda) @ W[K,N] + bias? + addsrc?), op = relu optional.
template<int NSUB>
__global__ __launch_bounds__(128)
void k_gemm(const float* __restrict__ A, int lda,
            const float* __restrict__ W,
            float* __restrict__ C,
            const float* __restrict__ bias,
            const float* __restrict__ addsrc,
            int M, int Kd, int Nd, int relu)
{
    const int lane = threadIdx.x & 31;
    const int wave = threadIdx.x >> 5;
    const int TN   = 16 * NSUB;
    const int ntn  = (Nd + TN - 1) / TN;
    const int ntm  = (M + 31) >> 5;          // 32 rows per tile (2 M-subtiles)
    const int tile = blockIdx.x * GEMM_WAVES + wave;
    if (tile >= ntm * ntn) return;           // wave-uniform: EXEC stays full
    const int tm   = tile / ntn;
    const int tn   = tile - tm * ntn;
    const int row0 = tm << 5;
    const int col0 = tn * TN;
    const bool lo  = lane < 16;
    const int  l15 = lane & 15;

    // Clamped rows/cols: invalid lanes only feed outputs masked at store.
    int r0 = row0 + l15;      if (r0 >= M) r0 = M - 1;
    int r1 = row0 + 16 + l15; if (r1 >= M) r1 = M - 1;
    int cB[NSUB];
#pragma unroll
    for (int t = 0; t < NSUB; ++t) {
        int c = col0 + t * 16 + l15;
        cB[t] = (c < Nd) ? c : (Nd - 1);
    }

    const int koff = lo ? 0 : 2;             // lanes 0-15: K,K+1 ; 16-31: K+2,K+3
    const unsigned uN     = (unsigned)Nd;
    const unsigned a0base = (unsigned)r0 * (unsigned)lda + (unsigned)koff;
    const unsigned a1base = (unsigned)r1 * (unsigned)lda + (unsigned)koff;
    const unsigned wbase  = (unsigned)koff * uN;

    v8f acc0[NSUB], acc1[NSUB];
    const v8f vz = {0.f,0.f,0.f,0.f,0.f,0.f,0.f,0.f};
#pragma unroll
    for (int t = 0; t < NSUB; ++t) { acc0[t] = vz; acc1[t] = vz; }

    const int Kmain = Kd & ~3;
    if (Kmain > 0) {
        // prologue: load fragments for k0 = 0
        v2f a0, a1; float bx[NSUB], by[NSUB];
        a0.x = A[a0base];     a0.y = A[a0base + 1];
        a1.x = A[a1base];     a1.y = A[a1base + 1];
#pragma unroll
        for (int t = 0; t < NSUB; ++t) {
            bx[t] = W[wbase + (unsigned)cB[t]];
            by[t] = W[wbase + uN + (unsigned)cB[t]];
        }
        // software-pipelined main loop: prefetch k0, compute k0-4
        for (int k0 = 4; k0 < Kmain; k0 += 4) {
            v2f na0, na1; float nbx[NSUB], nby[NSUB];
            const unsigned ao = (unsigned)k0;
            na0.x = A[a0base + ao]; na0.y = A[a0base + ao + 1];
            na1.x = A[a1base + ao]; na1.y = A[a1base + ao + 1];
            const unsigned wr = wbase + ao * uN;
#pragma unroll
            for (int t = 0; t < NSUB; ++t) {
                nbx[t] = W[wr + (unsigned)cB[t]];
                nby[t] = W[wr + uN + (unsigned)cB[t]];
            }
#pragma unroll
            for (int t = 0; t < NSUB; ++t) {
                v2f b; b.x = bx[t]; b.y = by[t];
                acc0[t] = WMMA_F32(a0, b, acc0[t]);
                acc1[t] = WMMA_F32(a1, b, acc1[t]);
            }
            a0 = na0; a1 = na1;
#pragma unroll
            for (int t = 0; t < NSUB; ++t) { bx[t] = nbx[t]; by[t] = nby[t]; }
        }
        // epilogue: compute last pipelined step
#pragma unroll
        for (int t = 0; t < NSUB; ++t) {
            v2f b; b.x = bx[t]; b.y = by[t];
            acc0[t] = WMMA_F32(a0, b, acc0[t]);
            acc1[t] = WMMA_F32(a1, b, acc1[t]);
        }
    }
    if (Kmain < Kd) {                        // K tail (Kd % 4), zero-filled
        const int ka = Kmain + koff;
        v2f a0t, a1t;
        a0t.x = (ka     < Kd) ? A[(unsigned)r0 * (unsigned)lda + (unsigned)ka]       : 0.f;
        a0t.y = (ka + 1 < Kd) ? A[(unsigned)r0 * (unsigned)lda + (unsigned)(ka + 1)] : 0.f;
        a1t.x = (ka     < Kd) ? A[(unsigned)r1 * (unsigned)lda + (unsigned)ka]       : 0.f;
        a1t.y = (ka + 1 < Kd) ? A[(unsigned)r1 * (unsigned)lda + (unsigned)(ka + 1)] : 0.f;
#pragma unroll
        for (int t = 0; t < NSUB; ++t) {
            v2f b;
            b.x = (ka     < Kd) ? W[(unsigned)ka * uN + (unsigned)cB[t]]       : 0.f;
            b.y = (ka + 1 < Kd) ? W[(unsigned)(ka + 1) * uN + (unsigned)cB[t]] : 0.f;
            acc0[t] = WMMA_F32(a0t, b, acc0[t]);
            acc1[t] = WMMA_F32(a1t, b, acc1[t]);
        }
    }

    // store: C layout vgpr i -> M = base + i (lanes 0-15) / base+8+i (16-31)
#pragma unroll
    for (int half = 0; half < 2; ++half) {
        const int mbase = row0 + half * 16 + (lo ? 0 : 8);
#pragma unroll
        for (int t = 0; t < NSUB; ++t) {
            const int c = col0 + t * 16 + l15;
            if (c < Nd) {
                const float bv = bias ? bias[c] : 0.f;
#pragma unroll
                for (int i = 0; i < 8; ++i) {
                    const int mrow = mbase + i;
                    if (mrow < M) {
                        const unsigned off = (unsigned)mrow * uN + (unsigned)c;
                        float vv = (half ? acc1[t][i] : acc0[t][i]) + bv;
                        if (addsrc) vv += addsrc[off];
                        if (relu)   vv = fmaxf(vv, 0.0f);
                        C[off] = vv;
                    }
                }
            }
        }
    }
}

// =========================== scalar kernels ================================

__global__ void k_extract_p(const float* __restrict__ pc, float* __restrict__ p, int npts)
{
    int tid = blockIdx.x * blockDim.x + threadIdx.x;
    if (tid >= npts * 3) return;
    int i = tid / 3, j = tid - i * 3;
    p[tid] = pc[(size_t)i * 9 + j];
}

// brute-force top-k (stable on ties -> matches lax.top_k tie-break)
template<int KK>
__global__ void k_knn(const float* __restrict__ Q, const float* __restrict__ R,
                      int Nq, int Nr, int* __restrict__ oidx, float* __restrict__ odist)
{
    int tid = blockIdx.x * blockDim.x + threadIdx.x;
    if (tid >= BATCH * Nq) return;
    int b = tid / Nq;
    const float* q = Q + (size_t)tid * 3;
    const float* r = R + (size_t)b * Nr * 3;
    const float qx = q[0], qy = q[1], qz = q[2];
    float bd[KK]; int bi[KK];
#pragma unroll
    for (int t = 0; t < KK; ++t) { bd[t] = 3.4e38f; bi[t] = 0; }
    for (int j = 0; j < Nr; ++j) {
        float dx = r[j*3+0]-qx, dy = r[j*3+1]-qy, dz = r[j*3+2]-qz;
        float d = dx*dx + dy*dy + dz*dz;
        if (d < bd[KK-1]) {
            bd[KK-1] = d; bi[KK-1] = j;
#pragma unroll
            for (int t = KK-1; t > 0; --t) {
                if (bd[t] < bd[t-1]) {
                    float td = bd[t]; bd[t] = bd[t-1]; bd[t-1] = td;
                    int   ti = bi[t]; bi[t] = bi[t-1]; bi[t-1] = ti;
                }
            }
        }
    }
    int* oi = oidx + (size_t)tid * KK;
#pragma unroll
    for (int t = 0; t < KK; ++t) oi[t] = bi[t];
    if (odist) {
        float* od = odist + (size_t)tid * KK;
#pragma unroll
        for (int t = 0; t < KK; ++t) od[t] = bd[t];
    }
}

// farthest point sampling: one block per batch element, sequential argmax
__global__ __launch_bounds__(512)
void k_fps(const float* __restrict__ P, int N, int m,
           int* __restrict__ idxs, float* __restrict__ dmin)
{
    __shared__ float sv[512];
    __shared__ int   si[512];
    __shared__ int   slast;
    const int b = blockIdx.x, t = threadIdx.x;
    const float* p = P + (size_t)b * N * 3;
    int*   id = idxs + (size_t)b * m;
    float* dm = dmin + (size_t)b * N;
    for (int i = t; i < N; i += 512) dm[i] = 1e10f;
    if (t == 0) { id[0] = 0; slast = 0; }
    __syncthreads();
    for (int it = 1; it < m; ++it) {
        const int last = slast;
        const float lx = p[last*3], ly = p[last*3+1], lz = p[last*3+2];
        float bv = -1.f; int bi = 0;
        for (int i = t; i < N; i += 512) {
            float dx = p[i*3]-lx, dy = p[i*3+1]-ly, dz = p[i*3+2]-lz;
            float d = dx*dx + dy*dy + dz*dz;
            float v = dm[i];
            if (d < v) { v = d; dm[i] = v; }
            if (v > bv) { bv = v; bi = i; }   // strictly-greater -> lowest idx
        }
        sv[t] = bv; si[t] = bi;
        __syncthreads();
        for (int s = 256; s > 0; s >>= 1) {
            if (t < s) {
                if (sv[t+s] > sv[t] || (sv[t+s] == sv[t] && si[t+s] < si[t])) {
                    sv[t] = sv[t+s]; si[t] = si[t+s];
                }
            }
            __syncthreads();
        }
        if (t == 0) { id[it] = si[0]; slast = si[0]; }
        __syncthreads();
    }
}

__global__ void k_gather_rows(const float* __restrict__ src, const int* __restrict__ idx,
                              float* __restrict__ dst, int mrows, int nsrc, int C)
{
    int tid = blockIdx.x * blockDim.x + threadIdx.x;
    if (tid >= BATCH * mrows * C) return;
    int c = tid % C, r = tid / C, b = r / mrows;
    dst[tid] = src[((size_t)b * nsrc + idx[r]) * C + c];
}

// feat[b,i,k,:] = [pnew - pold[nid], x[nid]]  (width C+3)
__global__ void k_build_feat(const float* __restrict__ pnew, const float* __restrict__ pold,
                             const float* __restrict__ x, const int* __restrict__ idx,
                             float* __restrict__ feat, int m, int Nold, int C, int K)
{
    const int W = C + 3;
    int tid = blockIdx.x * blockDim.x + threadIdx.x;
    if (tid >= BATCH * m * K * W) return;
    int c  = tid % W;
    int t2 = tid / W;
    int kk = t2 % K;
    int bi = t2 / K;
    int b  = bi / m;
    int nid = idx[(size_t)bi * K + kk];
    float val;
    if (c < 3) val = pnew[(size_t)bi * 3 + c] - pold[((size_t)b * Nold + nid) * 3 + c];
    else       val = x[((size_t)b * Nold + nid) * C + (c - 3)];
    feat[tid] = val;
}

__global__ void k_maxk(const float* __restrict__ t, float* __restrict__ out,
                       int m, int K, int C)
{
    int tid = blockIdx.x * blockDim.x + threadIdx.x;
    if (tid >= BATCH * m * C) return;
    int c = tid % C, bi = tid / C;
    size_t base = (size_t)bi * K * C + c;
    float mx = t[base];
    for (int kk = 1; kk < K; ++kk) mx = fmaxf(mx, t[base + (size_t)kk * C]);
    out[tid] = mx;
}

// pos = relu(rel @ pos1) @ pos2,  rel = p[n] - p[idx[n,k]]
__global__ void k_pos(const float* __restrict__ p, const int* __restrict__ idx,
                      const float* __restrict__ pos1, const float* __restrict__ pos2,
                      float* __restrict__ pos, int Nl, int K, int C)
{
    int tid = blockIdx.x * blockDim.x + threadIdx.x;
    if (tid >= BATCH * Nl * K * C) return;
    int c  = tid % C;
    int t2 = tid / C;
    int kk = t2 % K;
    int bn = t2 / K;
    int b  = bn / Nl;
    int nid = idx[(size_t)bn * K + kk];
    const float* pa = p + (size_t)bn * 3;
    const float* pb = p + ((size_t)b * Nl + nid) * 3;
    float r0 = pa[0]-pb[0], r1 = pa[1]-pb[1], r2 = pa[2]-pb[2];
    float out = 0.f;
#pragma unroll
    for (int j = 0; j < 3; ++j) {
        float h = r0 * pos1[0*3+j] + r1 * pos1[1*3+j] + r2 * pos1[2*3+j];
        h = fmaxf(h, 0.f);
        out += h * pos2[(size_t)j * C + c];
    }
    pos[tid] = out;
}

// u = relu(q - k[idx] + pos)
__global__ void k_u(const float* __restrict__ q, const float* __restrict__ kf,
                    const int* __restrict__ idx, const float* __restrict__ pos,
                    float* __restrict__ u, int Nl, int K, int C)
{
    int tid = blockIdx.x * blockDim.x + threadIdx.x;
    if (tid >= BATCH * Nl * K * C) return;
    int c  = tid % C;
    int t2 = tid / C;
    int kk = t2 % K;
    int bn = t2 / K;
    int b  = bn / Nl;
    int nid = idx[(size_t)bn * K + kk];
    float val = q[(size_t)bn * C + c]
              - kf[((size_t)b * Nl + nid) * C + c]
              + pos[tid];
    u[tid] = fmaxf(val, 0.f);
}

// agg = sum_k softmax_k(s) * (v[idx] + pos)
__global__ void k_softagg(const float* __restrict__ s, const float* __restrict__ v,
                          const float* __restrict__ pos, const int* __restrict__ idx,
                          float* __restrict__ agg, int Nl, int K, int C)
{
    int tid = blockIdx.x * blockDim.x + threadIdx.x;
    if (tid >= BATCH * Nl * C) return;
    int c  = tid % C;
    int bn = tid / C;
    int b  = bn / Nl;
    size_t base = (size_t)bn * K * C + c;
    const int* id = idx + (size_t)bn * K;
    float mx = -3.4e38f;
    for (int kk = 0; kk < K; ++kk) mx = fmaxf(mx, s[base + (size_t)kk * C]);
    float den = 0.f, num = 0.f;
    for (int kk = 0; kk < K; ++kk) {
        float e = expf(s[base + (size_t)kk * C] - mx);
        int nid = id[kk];
        num += e * (v[((size_t)b * Nl + nid) * C + c] + pos[base + (size_t)kk * C]);
        den += e;
    }
    agg[tid] = num / den;
}

// y = relu(sum_t iw_t * g[ni_t] + skip), iw_t = (1/(d_t+1e-8)) normalized
__global__ void k_upcombine(const float* __restrict__ g, const float* __restrict__ skip,
                            const int* __restrict__ ni, const float* __restrict__ nd,
                            float* __restrict__ out, int Nf, int Nc, int C)
{
    int tid = blockIdx.x * blockDim.x + threadIdx.x;
    if (tid >= BATCH * Nf * C) return;
    int c = tid % C, r = tid / C, b = r / Nf;
    size_t r3 = (size_t)r * 3;
    float w0 = 1.f / (nd[r3+0] + 1e-8f);
    float w1 = 1.f / (nd[r3+1] + 1e-8f);
    float w2 = 1.f / (nd[r3+2] + 1e-8f);
    float sw = w0 + w1 + w2;
    const float* gb = g + (size_t)b * Nc * C;
    float val = (w0 * gb[(size_t)ni[r3+0] * C + c] +
                 w1 * gb[(size_t)ni[r3+1] * C + c] +
                 w2 * gb[(size_t)ni[r3+2] * C + c]) / sw
              + skip[tid];
    out[tid] = fmaxf(val, 0.f);
}

__global__ void k_logsoftmax(const float* __restrict__ x, float* __restrict__ out,
                             int rows, int D)
{
    int r = blockIdx.x * blockDim.x + threadIdx.x;
    if (r >= rows) return;
    const float* v = x + (size_t)r * D;
    float* o = out + (size_t)r * D;
    float mx = v[0];
    for (int i = 1; i < D; ++i) mx = fmaxf(mx, v[i]);
    float s = 0.f;
    for (int i = 0; i < D; ++i) s += expf(v[i] - mx);
    float lse = mx + logf(s);
    for (int i = 0; i < D; ++i) o[i] = v[i] - lse;
}

__global__ void k_normparam(const float* __restrict__ x, float* __restrict__ out, int rows)
{
    int r = blockIdx.x * blockDim.x + threadIdx.x;
    if (r >= rows) return;
    const float* v = x + (size_t)r * 22;
    float* o = out + (size_t)r * 22;
    for (int i = 0; i < 22; ++i) o[i] = v[i];
    const int segs[3] = {4, 8, 15};
#pragma unroll
    for (int si = 0; si < 3; ++si) {
        int s = segs[si];
        float n = sqrtf(v[s]*v[s] + v[s+1]*v[s+1] + v[s+2]*v[s+2]);
        float inv = 1.f / (n + 1e-12f);
        o[s] = v[s]*inv; o[s+1] = v[s+1]*inv; o[s+2] = v[s+2]*inv;
    }
}

// ============================ host orchestration ===========================

static inline dim3 ewgrid(size_t n) { return dim3((unsigned)((n + 255) / 256)); }

static inline void gemm(hipStream_t s, const float* A, int lda, const float* W,
                        float* C, const float* bias, const float* add,
                        int M, int K, int N, int relu)
{
    const int ntm = (M + 31) / 32;
    if (N >= 64) {
        const int tiles = ntm * ((N + 63) / 64);
        k_gemm<4><<<dim3((tiles + GEMM_WAVES - 1) / GEMM_WAVES), dim3(128), 0, s>>>(
            A, lda, W, C, bias, add, M, K, N, relu);
    } else if (N > 16) {
        const int tiles = ntm * ((N + 31) / 32);
        k_gemm<2><<<dim3((tiles + GEMM_WAVES - 1) / GEMM_WAVES), dim3(128), 0, s>>>(
            A, lda, W, C, bias, add, M, K, N, relu);
    } else {
        const int tiles = ntm * ((N + 15) / 16);
        k_gemm<1><<<dim3((tiles + GEMM_WAVES - 1) / GEMM_WAVES), dim3(128), 0, s>>>(
            A, lda, W, C, bias, add, M, K, N, relu);
    }
}

extern "C" void kernel_launch(void* const* d_in, const int* in_sizes, int n_in,
                              void* d_out, int out_size, void* d_ws, size_t ws_size,
                              hipStream_t stream)
{
    (void)in_sizes; (void)n_in; (void)out_size; (void)ws_size;
    const float* pc = (const float*)d_in[0];
    float* out = (float*)d_out;
    auto P = [&](int i) -> const float* { return (const float*)d_in[i]; };

    static const int NPv[5]  = {8192, 2048, 512, 128, 32};
    static const int CHv[5]  = {32, 64, 128, 256, 512};
    static const int KSv[5]  = {8, 16, 16, 16, 16};
    static const int NBLK[5] = {2, 3, 4, 6, 3};
    static const int ENC_CUM[5] = {0, 2, 5, 9, 15};

    // --- parameter leaf indices (jax tree order: sorted dict keys) ---
    // 1..90: dec (5 lvls x 2 blks x 9) | 91,92: dec_mlp | 93..96: down
    // 97..258: enc (18 blks x 9) | 259..261: head_emb (b2,w1,w2)
    // 262..264: head_param | 265..267: head_prim | 268,269: in_mlp | 270..277: up
    auto decB = [&](int l, int b) { return 1 + (l * 2 + b) * 9; };
    auto encB = [&](int l, int b) { return 97 + (ENC_CUM[l] + b) * 9; };

    // --- workspace carve (bump allocator) ---
    char* wp = (char*)d_ws;
    auto af = [&](size_t n) -> float* {
        float* q = (float*)wp; wp += ((n * 4 + 255) & ~(size_t)255); return q; };
    auto ai = [&](size_t n) -> int* {
        int* q = (int*)wp;     wp += ((n * 4 + 255) & ~(size_t)255); return q; };

    float *p_[5], *xs_[5], *yb_[5]; int* kn_[5];
    for (int l = 0; l < 5; ++l) {
        p_[l]  = af((size_t)BATCH * NPv[l] * 3);
        xs_[l] = af((size_t)BATCH * NPv[l] * CHv[l]);
        yb_[l] = af((size_t)BATCH * NPv[l] * CHv[l]);
        kn_[l] = ai((size_t)BATCH * NPv[l] * KSv[l]);
    }
    float* T0 = af(524288);  float* T1 = af(524288);
    float* T2 = af(524288);  float* T3 = af(524288);
    float* S0 = af(4194304); float* S1 = af(4194304);
    float* S2 = af(4194304); float* S3 = af(4194304);
    int*   dknn = ai(65536);
    int*   ni3  = ai((size_t)BATCH * 8192 * 3);
    float* nd3  = af((size_t)BATCH * 8192 * 3);
    float* dmin = af((size_t)BATCH * 8192);
    int*   fpsi = ai((size_t)BATCH * 2048);

    // one full transformer block, in-place on x
    auto run_block = [&](float* x, const float* p, const int* idx,
                         int Nl, int C, int K, int wb) {
        const float *a1 = P(wb+0), *a2 = P(wb+1), *pos1 = P(wb+2), *pos2 = P(wb+3);
        const float *w_in = P(wb+4), *w_out = P(wb+5);
        const float *wk = P(wb+6), *wq = P(wb+7), *wv = P(wb+8);
        const int rows = BATCH * Nl;
        const size_t nkc = (size_t)rows * K * C;
        gemm(stream, x,  C, w_in, T0, nullptr, nullptr, rows, C, C, 1); // y=relu(x@w_in)
        gemm(stream, T0, C, wq,   T1, nullptr, nullptr, rows, C, C, 0); // q
        gemm(stream, T0, C, wk,   T2, nullptr, nullptr, rows, C, C, 0); // k
        gemm(stream, T0, C, wv,   T3, nullptr, nullptr, rows, C, C, 0); // v
        k_pos<<<ewgrid(nkc), 256, 0, stream>>>(p, idx, pos1, pos2, S0, Nl, K, C);
        k_u  <<<ewgrid(nkc), 256, 0, stream>>>(T1, T2, idx, S0, S1, Nl, K, C);
        gemm(stream, S1, C, a1, S2, nullptr, nullptr, rows * K, C, C, 1); // relu(u@a1)
        gemm(stream, S2, C, a2, S3, nullptr, nullptr, rows * K, C, C, 0); // s
        k_softagg<<<ewgrid((size_t)rows * C), 256, 0, stream>>>(S3, T3, S0, idx, T1, Nl, K, C);
        gemm(stream, T1, C, w_out, x, nullptr, x, rows, C, C, 1); // x=relu(x+agg@w_out)
    };

    // ---------------- encoder ----------------
    k_extract_p<<<ewgrid((size_t)BATCH * 8192 * 3), 256, 0, stream>>>(pc, p_[0], BATCH * 8192);
    gemm(stream, pc + 3, 9, P(268), T0, nullptr, nullptr, BATCH * 8192, 6, 32, 1);
    gemm(stream, T0, 32, P(269), xs_[0], nullptr, nullptr, BATCH * 8192, 32, 32, 1);

    for (int l = 0; l < 5; ++l) {
        if (l > 0) {
            const int Np = NPv[l-1], m = NPv[l], Cp = CHv[l-1], Cn = CHv[l], K = KSv[l];
            k_fps<<<dim3(BATCH), dim3(512), 0, stream>>>(p_[l-1], Np, m, fpsi, dmin);
            k_gather_rows<<<ewgrid((size_t)BATCH * m * 3), 256, 0, stream>>>(
                p_[l-1], fpsi, p_[l], m, Np, 3);
            k_knn<16><<<ewgrid((size_t)BATCH * m), 128, 0, stream>>>(
                p_[l], p_[l-1], m, Np, dknn, nullptr);
            k_build_feat<<<ewgrid((size_t)BATCH * m * K * (Cp + 3)), 256, 0, stream>>>(
                p_[l], p_[l-1], xs_[l-1], dknn, S0, m, Np, Cp, K);
            gemm(stream, S0, Cp + 3, P(92 + l), S1, nullptr, nullptr,
                 BATCH * m * K, Cp + 3, Cn, 1);
            k_maxk<<<ewgrid((size_t)BATCH * m * Cn), 256, 0, stream>>>(S1, xs_[l], m, K, Cn);
        }
        if (l == 0)
            k_knn<8><<<ewgrid((size_t)BATCH * NPv[0]), 128, 0, stream>>>(
                p_[0], p_[0], NPv[0], NPv[0], kn_[0], nullptr);
        else
            k_knn<16><<<ewgrid((size_t)BATCH * NPv[l]), 128, 0, stream>>>(
                p_[l], p_[l], NPv[l], NPv[l], kn_[l], nullptr);
        for (int b = 0; b < NBLK[l]; ++b)
            run_block(xs_[l], p_[l], kn_[l], NPv[l], CHv[l], KSv[l], encB(l, b));
    }

    // ---------------- decoder ----------------
    gemm(stream, xs_[4], 512, P(91), T0, nullptr, nullptr, BATCH * 32, 512, 512, 1);
    gemm(stream, T0, 512, P(92), yb_[4], nullptr, nullptr, BATCH * 32, 512, 512, 1);
    for (int b = 0; b < 2; ++b)
        run_block(yb_[4], p_[4], kn_[4], NPv[4], CHv[4], KSv[4], decB(4, b));

    for (int l = 3; l >= 0; --l) {
        const int Nf = NPv[l], Nc = NPv[l+1], Cf = CHv[l], Cc = CHv[l+1];
        k_knn<3><<<ewgrid((size_t)BATCH * Nf), 128, 0, stream>>>(
            p_[l], p_[l+1], Nf, Nc, ni3, nd3);
        gemm(stream, yb_[l+1], Cc, P(270 + 2 * l), T0, nullptr, nullptr, BATCH * Nc, Cc, Cf, 0);
        gemm(stream, xs_[l],   Cf, P(271 + 2 * l), T1, nullptr, nullptr, BATCH * Nf, Cf, Cf, 0);
        k_upcombine<<<ewgrid((size_t)BATCH * Nf * Cf), 256, 0, stream>>>(
            T0, T1, ni3, nd3, yb_[l], Nf, Nc, Cf);
        for (int b = 0; b < 2; ++b)
            run_block(yb_[l], p_[l], kn_[l], NPv[l], CHv[l], KSv[l], decB(l, b));
    }

    // ---------------- heads ----------------
    const int rows = BATCH * 8192;
    float* y0 = yb_[0];
    // emb = relu(y@w1)@w2 + b2 -> out[0 : rows*128]
    gemm(stream, y0, 32, P(260), T0, nullptr, nullptr, rows, 32, 32, 1);
    gemm(stream, T0, 32, P(261), out, P(259), nullptr, rows, 32, 128, 0);
    // prim = log_softmax(relu(y@w1)@w2 + b2) -> out[rows*128 : +rows*10]
    gemm(stream, y0, 32, P(266), T1, nullptr, nullptr, rows, 32, 32, 1);
    gemm(stream, T1, 32, P(267), T2, P(265), nullptr, rows, 32, 10, 0);
    k_logsoftmax<<<ewgrid(rows), 256, 0, stream>>>(T2, out + (size_t)rows * 128, rows, 10);
    // param head -> out[rows*138 : +rows*22]
    gemm(stream, y0, 32, P(263), T0, nullptr, nullptr, rows, 32, 32, 1);
    gemm(stream, T0, 32, P(264), T1, P(262), nullptr, rows, 32, 22, 0);
    k_normparam<<<ewgrid(rows), 256, 0, stream>>>(T1, out + (size_t)rows * 138, rows);
}